// LinearAttention_81372450390337
// MI455X (gfx1250) — compile-verified
//
#include <hip/hip_runtime.h>
#include <math.h>

typedef float v2f __attribute__((ext_vector_type(2)));
typedef float v8f __attribute__((ext_vector_type(8)));

#define T_DIM 2048
#define B_DIM 32
#define H_DIM 1024

// ---------------------------------------------------------------------------
// Kernel A: v[b,n] = sum_k hidden[b,k] * W[k,n]      (v = hidden @ W)
// Tiny fp32 GEMM (M=32, N=1024, K=1024) done with V_WMMA_F32_16X16X4_F32.
// One wave per 16x16 output tile; k-loop steps 4.
// A-frag (16x4 f32): lane%16 = M row; lanes 0-15 hold K={k,k+1} in {V0,V1},
//                    lanes 16-31 hold K={k+2,k+3}.
// B-frag (4x16 f32): lane%16 = N col; same K split by lane half.
// D (16x16 f32): VGPR j -> M=j (lanes 0-15) / M=j+8 (lanes 16-31), N=lane%16.
// ---------------------------------------------------------------------------
__global__ __launch_bounds__(32) void attn_vproj_wmma(
    const float* __restrict__ hidden,  // [B, H]
    const float* __restrict__ W,       // [H, H] row-major (W[k][n])
    float* __restrict__ v)             // [B, H]
{
    const int lane   = threadIdx.x;        // 0..31, full wave (EXEC all 1s)
    const int nBase  = blockIdx.x * 16;    // N tile
    const int mBase  = blockIdx.y * 16;    // M tile
    const int laneLo = lane & 15;
    const int half   = lane >> 4;          // 0: lanes 0-15, 1: lanes 16-31

    const float* aRow = hidden + (mBase + laneLo) * H_DIM;
    const float* bCol = W + nBase + laneLo;

    v8f acc = {};
    for (int k = 0; k < H_DIM; k += 4) {
        const int ka = k + 2 * half;
        v2f a, bm;
        a.x  = aRow[ka];
        a.y  = aRow[ka + 1];
        bm.x = bCol[(size_t)ka * H_DIM];
        bm.y = bCol[(size_t)(ka + 1) * H_DIM];
        acc = __builtin_amdgcn_wmma_f32_16x16x4_f32(
            /*neg_a=*/false, a, /*neg_b=*/false, bm,
            /*c_mod=*/(short)0, acc, /*reuse_a=*/false, /*reuse_b=*/false);
    }

#pragma unroll
    for (int j = 0; j < 8; ++j) {
        const int m = mBase + j + 8 * half;
        v[(size_t)m * H_DIM + nBase + laneLo] = acc[j];
    }
}

// ---------------------------------------------------------------------------
// Kernel B: scores[b,t] = dot(v[b,:], enc[t,b,:])
// Memory-bound: streams 256 MB of enc exactly once (~11 us @ 23.3 TB/s).
// One wave per (t,b) row: 32 lanes x 8 float4 loads = 1024 floats.
// v (128 KB total) stays resident in L2.
// ---------------------------------------------------------------------------
__global__ __launch_bounds__(256) void attn_scores(
    const float* __restrict__ enc,     // [T, B, H]
    const float* __restrict__ v,       // [B, H]
    float* __restrict__ scores)        // [B, T]
{
    const int lane = threadIdx.x & 31;
    const int wib  = threadIdx.x >> 5;
    const long w   = (long)blockIdx.x * 8 + wib;   // row id in [0, T*B)
    const int b    = (int)(w % B_DIM);
    const int t    = (int)(w / B_DIM);

    const float4* er = (const float4*)(enc + ((long)t * B_DIM + b) * H_DIM);
    const float4* vr = (const float4*)(v + (long)b * H_DIM);

    float s = 0.f;
#pragma unroll
    for (int it = 0; it < 8; ++it) {
        const float4 e = er[it * 32 + lane];
        const float4 q = vr[it * 32 + lane];
        s += e.x * q.x + e.y * q.y + e.z * q.z + e.w * q.w;
    }
    // wave32 cross-lane reduction
#pragma unroll
    for (int off = 16; off >= 1; off >>= 1)
        s += __shfl_xor(s, off, 32);

    if (lane == 0)
        scores[(long)b * T_DIM + t] = s;
}

// ---------------------------------------------------------------------------
// Kernel C: out[b,0,:] = softmax(scores[b,:]) over T.
// Note: the hidden.bias constant c[b] is omitted upstream — softmax is
// invariant to a per-row additive constant, so the result is identical.
// ---------------------------------------------------------------------------
__global__ __launch_bounds__(256) void attn_softmax(
    const float* __restrict__ scores,  // [B, T]
    float* __restrict__ out)           // [B, 1, T]
{
    const int b   = blockIdx.x;
    const int tid = threadIdx.x;
    const float* row = scores + (long)b * T_DIM;
    __shared__ float red[256];

    float m = -INFINITY;
    for (int t = tid; t < T_DIM; t += 256) m = fmaxf(m, row[t]);
    red[tid] = m;
    __syncthreads();
    for (int s = 128; s > 0; s >>= 1) {
        if (tid < s) red[tid] = fmaxf(red[tid], red[tid + s]);
        __syncthreads();
    }
    m = red[0];
    __syncthreads();

    float sum = 0.f;
    for (int t = tid; t < T_DIM; t += 256) sum += __expf(row[t] - m);
    red[tid] = sum;
    __syncthreads();
    for (int s = 128; s > 0; s >>= 1) {
        if (tid < s) red[tid] += red[tid + s];
        __syncthreads();
    }
    const float inv = 1.f / red[0];

    for (int t = tid; t < T_DIM; t += 256)
        out[(long)b * T_DIM + t] = __expf(row[t] - m) * inv;
}

// ---------------------------------------------------------------------------
extern "C" void kernel_launch(void* const* d_in, const int* in_sizes, int n_in,
                              void* d_out, int out_size, void* d_ws, size_t ws_size,
                              hipStream_t stream) {
    const float* hidden = (const float*)d_in[0];  // [1, B, H]
    const float* enc    = (const float*)d_in[1];  // [T, B, H]
    const float* W      = (const float*)d_in[2];  // [H, H]
    // d_in[3] = bias: contribution is constant per softmax row -> cancels.

    float* v      = (float*)d_ws;                 // B*H floats   (128 KB)
    float* scores = v + (size_t)B_DIM * H_DIM;    // B*T floats   (256 KB)

    attn_vproj_wmma<<<dim3(H_DIM / 16, B_DIM / 16), 32, 0, stream>>>(hidden, W, v);
    attn_scores<<<(T_DIM * B_DIM) / 8, 256, 0, stream>>>(enc, v, scores);
    attn_softmax<<<B_DIM, 256, 0, stream>>>(scores, (float*)d_out);
}